// HashEmbedding_2920577761815
// MI455X (gfx1250) — compile-verified
//
#include <hip/hip_runtime.h>
#include <cmath>

// InstantNGP hash-grid embed, MI455X (gfx1250, wave32).
// One thread per (point, level). 256-thread block = 16 points x 16 levels.
// Lanes 0..15 of a point write 16 contiguous float2 (128B) -> coalesced.

#define N_LEVELS   16
#define LOG2_T     19
#define TABLE_SIZE (1u << LOG2_T)
#define TMASK      (TABLE_SIZE - 1u)

struct ResArg { float r[N_LEVELS]; };

__global__ __launch_bounds__(256)
void hashgrid_embed_kernel(const float*  __restrict__ x,
                           const float*  __restrict__ emb,
                           float2*       __restrict__ out,
                           ResArg rc, int npts)
{
    __shared__ float sx[48];                      // 16 points * 3 coords
    const int t    = threadIdx.x;
    const int base = blockIdx.x * 16;             // first point of this block

    // Stage this block's coordinates through LDS (coalesced 48-float read).
    if (t < 48 && (base * 3 + t) < npts * 3)
        sx[t] = x[base * 3 + t];
    // Speculative prefetch of next block's coords (gfx1250 global_prefetch_b8).
    if (t == 48 && (base + 16) < npts)
        __builtin_prefetch(&x[(size_t)(base + 16) * 3], 0, 0);
    __syncthreads();

    const int p = t >> 4;                         // point within block
    const int l = t & 15;                         // level
    const int b = base + p;
    if (b >= npts) return;

    const float xx = sx[p * 3 + 0];
    const float xy = sx[p * 3 + 1];
    const float xz = sx[p * 3 + 2];

    // Reference math, op-for-op in f32.
    const float res  = rc.r[l];
    const float cell = 2.0f / res;

    const float f0 = floorf((xx + 1.0f) / cell);
    const float f1 = floorf((xy + 1.0f) / cell);
    const float f2 = floorf((xz + 1.0f) / cell);

    const float t0 = (xx - (f0 * cell - 1.0f)) / cell;
    const float t1 = (xy - (f1 * cell - 1.0f)) / cell;
    const float t2 = (xz - (f2 * cell - 1.0f)) / cell;

    const unsigned i0 = (unsigned)f0;
    const unsigned i1 = (unsigned)f1;
    const unsigned i2 = (unsigned)f2;

    const float w0[2] = { 1.0f - t0, t0 };
    const float w1[2] = { 1.0f - t1, t1 };
    const float w2[2] = { 1.0f - t2, t2 };

    const float* tab = emb + ((size_t)l << (LOG2_T + 1));   // l * T * 2 floats

    // Phase 1: all 8 corner hashes.
    unsigned h[8];
#pragma unroll
    for (int c = 0; c < 8; ++c) {
        const unsigned v0 = i0 + ( c       & 1);
        const unsigned v1 = i1 + ((c >> 1) & 1);
        const unsigned v2 = i2 + ((c >> 2) & 1);
        h[c] = (v0 ^ (v1 * 2654435761u) ^ (v2 * 805459861u)) & TMASK;
    }

    // Phase 2: issue all 8 b64 gathers back-to-back (L2-resident tables).
    float2 g[8];
#pragma unroll
    for (int c = 0; c < 8; ++c)
        g[c] = *(const float2*)(tab + ((size_t)h[c] << 1));

    // Phase 3: trilinear-weighted accumulation.
    float a0 = 0.0f, a1 = 0.0f;
#pragma unroll
    for (int c = 0; c < 8; ++c) {
        const float w = w0[c & 1] * w1[(c >> 1) & 1] * w2[(c >> 2) & 1];
        a0 = fmaf(w, g[c].x, a0);
        a1 = fmaf(w, g[c].y, a1);
    }

    // out[b][l*2 + f]: 16 consecutive lanes -> 128B contiguous store.
    out[(size_t)b * 16 + l] = make_float2(a0, a1);
}

extern "C" void kernel_launch(void* const* d_in, const int* in_sizes, int n_in,
                              void* d_out, int out_size, void* d_ws, size_t ws_size,
                              hipStream_t stream)
{
    const float* x   = (const float*)d_in[0];   // (B,3)  f32
    const float* emb = (const float*)d_in[1];   // (16, 2^19, 2) f32
    float2* out = (float2*)d_out;               // (B, 32) f32 viewed as (B,16) float2

    const int npts = in_sizes[0] / 3;

    // Replicate numpy's resolution computation exactly (f64 exp/log/pow).
    ResArg rc;
    const double factor = std::exp((std::log(512.0) - std::log(16.0)) / 15.0);
    for (int i = 0; i < N_LEVELS; ++i)
        rc.r[i] = (float)std::floor(16.0 * std::pow(factor, (double)i));

    const int blocks = (npts + 15) / 16;        // 16 points per block
    hashgrid_embed_kernel<<<blocks, 256, 0, stream>>>(x, emb, out, rc, npts);
}